// VQC_Combined_8469675508244
// MI455X (gfx1250) — compile-verified
//
#include <hip/hip_runtime.h>

// ---------------------------------------------------------------------------
// VQC collapses to a 16x4 lookup table:
//   out[b][w] = table[idx(b)][w],  idx = x1_0*8 + x1_1*4 + x2_0*2 + x2_1
// Kernel 1 (one wave32): build U (16x16 complex unitary) in WMMA C-layout via
//   register butterflies, then table = Z_pad x |U|^2 via v_wmma_f32_16x16x4_f32.
//   Trig uses __sinf/__cosf (v_sin_f32/v_cos_f32) -- weights are O(1), the
//   hardware-range path is accurate there and avoids the huge OCML reduction.
// Kernel 2: memory-bound coalesced lookup (b128 NT loads/stores, LDS table).
// ---------------------------------------------------------------------------

typedef float v2f __attribute__((ext_vector_type(2)));
typedef float v8f __attribute__((ext_vector_type(8)));
typedef float f4  __attribute__((ext_vector_type(4)));
typedef int   i4  __attribute__((ext_vector_type(4)));

// ---------------------------------------------------------------------------
// Setup: one wave computes the 16x4 expectation table.
// U layout (WMMA C/D layout): lane L, reg r  ->  row M = r + 8*(L>=16),
// col N = L & 15.  Wire w acts on bit (3-w) of M.
// ---------------------------------------------------------------------------
__global__ __launch_bounds__(32)
void vqc_build_table(const float* __restrict__ wa, const float* __restrict__ wb,
                     const float* __restrict__ sa, const float* __restrict__ sb,
                     float* __restrict__ table)
{
    const int L = threadIdx.x;   // lane 0..31
    const int h = L >> 4;        // lane half == bit3 of row index M
    const int N = L & 15;        // column index in C-layout

    // U = I
    float ur[8], ui[8];
#pragma unroll
    for (int r = 0; r < 8; ++r) {
        ur[r] = ((r + 8 * h) == N) ? 1.f : 0.f;
        ui[r] = 0.f;
    }

    // conditional swap (upper lane half only) and full swap of register rows
    auto cswap = [&](int rA, int rB) {
        float arv = ur[rA], brv = ur[rB], aiv = ui[rA], biv = ui[rB];
        ur[rA] = h ? brv : arv;  ur[rB] = h ? arv : brv;
        ui[rA] = h ? biv : aiv;  ui[rB] = h ? aiv : biv;
    };
    auto fswap = [&](int rA, int rB) {
        float t = ur[rA]; ur[rA] = ur[rB]; ur[rB] = t;
        t = ui[rA]; ui[rA] = ui[rB]; ui[rB] = t;
    };

    const float* wsrc[4] = { wa, wa + 3, wb, wb + 3 };  // per-wire, stride 6/layer

#pragma unroll
    for (int layer = 0; layer < 4; ++layer) {
#pragma unroll
        for (int wire = 0; wire < 4; ++wire) {
            const float* p = wsrc[wire] + layer * 6;
            float phi = p[0], theta = p[1], omega = p[2];
            // fast hw trig: v_sin_f32 / v_cos_f32 (args are O(1) radians)
            float c  = __cosf(0.5f * theta), s = __sinf(0.5f * theta);
            float av = 0.5f * (phi + omega), bv = 0.5f * (phi - omega);
            float ca = __cosf(av), sna = __sinf(av);
            float cb = __cosf(bv), snb = __sinf(bv);
            // Rot = [[e^-ia c, -e^ib s], [e^-ib s, e^ia c]]
            float re00 =  c * ca, im00 = -c * sna;
            float re01 = -s * cb, im01 = -s * snb;
            float re10 =  s * cb, im10 = -s * snb;
            float re11 =  c * ca, im11 =  c * sna;

            if (wire == 0) {
                // bit3 == lane half: butterfly across lane halves
                float csr = h ? re11 : re00, csi = h ? im11 : im00;
                float cor = h ? re10 : re01, coi = h ? im10 : im01;
#pragma unroll
                for (int r = 0; r < 8; ++r) {
                    float mr = ur[r], mi = ui[r];
                    float qr = __shfl_xor(mr, 16, 32);
                    float qi = __shfl_xor(mi, 16, 32);
                    ur[r] = csr * mr - csi * mi + cor * qr - coi * qi;
                    ui[r] = csr * mi + csi * mr + cor * qi + coi * qr;
                }
            } else {
                int bit = 4 >> (wire - 1);   // wire1->4, wire2->2, wire3->1
#pragma unroll
                for (int r0 = 0; r0 < 8; ++r0) {
                    if (r0 & bit) continue;
                    int r1 = r0 | bit;
                    float xr0 = ur[r0], xi0 = ui[r0];
                    float xr1 = ur[r1], xi1 = ui[r1];
                    ur[r0] = re00*xr0 - im00*xi0 + re01*xr1 - im01*xi1;
                    ui[r0] = re00*xi0 + im00*xr0 + re01*xi1 + im01*xr1;
                    ur[r1] = re10*xr0 - im10*xi0 + re11*xr1 - im11*xi1;
                    ui[r1] = re10*xi0 + im10*xr0 + re11*xi1 + im11*xr1;
                }
            }
        }

        // CNOT(0,1): rows with bit3=1 (upper half): r <-> r^4
        cswap(0, 4); cswap(1, 5); cswap(2, 6); cswap(3, 7);
        // CNOT(2,3): rows with bit1=1: r <-> r^1  (uniform over halves)
        fswap(2, 3); fswap(6, 7);
        // CNOT(0,2): upper half: r <-> r^2
        cswap(0, 2); cswap(1, 3); cswap(4, 6); cswap(5, 7);
        // CNOT(0,3): upper half: r <-> r^1
        cswap(0, 1); cswap(2, 3); cswap(4, 5); cswap(6, 7);
    }

    // P[s][idx] = |U[s][idx]|^2, still in C-layout
    float prb[8];
#pragma unroll
    for (int r = 0; r < 8; ++r) prb[r] = ur[r] * ur[r] + ui[r] * ui[r];

    // T = Z_pad(16x16) x P(16x16) with four v_wmma_f32_16x16x4_f32 (K chunks)
    // A layout (documented): lane L holds M = L&15; reg j holds K = 4k + 2h + j
    // B layout (assumed mirror): lane L holds N = L&15; reg j holds K = 4k+2h+j
    v8f acc = {};
#pragma unroll
    for (int k = 0; k < 4; ++k) {
        v2f A, B;
#pragma unroll
        for (int j = 0; j < 2; ++j) {
            int K = 4 * k + 2 * h + j;
            int M = N;                       // Z row (wire), valid for M<4
            int sh = (M < 4) ? (3 - M) : 0;
            float z = 1.f - 2.f * (float)((K >> sh) & 1);
            A[j] = (M < 4) ? z : 0.f;
        }
        // P[K][N] resides at lane N + 16*(K>=8), register K&7 (K in [4k,4k+3])
        int srcLane = N + ((k >= 2) ? 16 : 0);
#pragma unroll
        for (int j = 0; j < 2; ++j) {
            float vlo = __shfl(prb[(4 * k + j)     & 7], srcLane, 32);
            float vhi = __shfl(prb[(4 * k + j + 2) & 7], srcLane, 32);
            B[j] = h ? vhi : vlo;
        }
        acc = __builtin_amdgcn_wmma_f32_16x16x4_f32(
                  false, A, false, B, (short)0, acc, false, false);
    }

    // D: lane L(<16), reg m  ->  T[w=m][idx=L];  q = (ev+1)/2 * scale[w]
    float sc[4] = { sa[0], sa[1], sb[0], sb[1] };
    if (L < 16) {
#pragma unroll
        for (int m = 0; m < 4; ++m)
            table[L * 4 + m] = (acc[m] + 1.f) * 0.5f * sc[m];
    }
}

// ---------------------------------------------------------------------------
// Batch lookup: each thread handles 2 batch elements (one b128 load per input,
// two ds_load_b128 table gathers, two b128 NT stores). ~16.8 MB total traffic
// -> ~0.72 us at 23.3 TB/s; this kernel is the entire runtime budget.
// ---------------------------------------------------------------------------
__global__ __launch_bounds__(256)
void vqc_lookup(const int* __restrict__ x1, const int* __restrict__ x2,
                const float* __restrict__ table, float* __restrict__ out,
                int npairs)
{
    __shared__ __align__(16) float tbl[64];
    int tid = threadIdx.x;
    if (tid < 64) tbl[tid] = table[tid];
    __syncthreads();

    int g = blockIdx.x * blockDim.x + tid;
    if (g >= npairs) return;

    i4 a = __builtin_nontemporal_load((const i4*)x1 + g);  // elems 2g, 2g+1 of x1
    i4 b = __builtin_nontemporal_load((const i4*)x2 + g);
    int idx0 = a.x * 8 + a.y * 4 + b.x * 2 + b.y;
    int idx1 = a.z * 8 + a.w * 4 + b.z * 2 + b.w;

    f4 r0 = *(const f4*)&tbl[idx0 * 4];
    f4 r1 = *(const f4*)&tbl[idx1 * 4];
    __builtin_nontemporal_store(r0, (f4*)out + 2 * g);
    __builtin_nontemporal_store(r1, (f4*)out + 2 * g + 1);
}

// ---------------------------------------------------------------------------
extern "C" void kernel_launch(void* const* d_in, const int* in_sizes, int n_in,
                              void* d_out, int out_size, void* d_ws, size_t ws_size,
                              hipStream_t stream)
{
    const int*   x1 = (const int*)d_in[0];
    const int*   x2 = (const int*)d_in[1];
    const float* wa = (const float*)d_in[2];
    const float* wb = (const float*)d_in[3];
    const float* sa = (const float*)d_in[4];
    const float* sb = (const float*)d_in[5];
    float* table = (float*)d_ws;          // 64 floats of scratch
    float* out   = (float*)d_out;

    int batch  = in_sizes[0] / 2;         // x1 is (BATCH, 2)
    int npairs = batch / 2;               // 2 elements per thread

    vqc_build_table<<<1, 32, 0, stream>>>(wa, wb, sa, sb, table);

    int threads = 256;
    int blocks  = (npairs + threads - 1) / threads;
    vqc_lookup<<<blocks, threads, 0, stream>>>(x1, x2, table, out, npairs);
}